// soft_cldice_loss_45432164057837
// MI455X (gfx1250) — compile-verified
//
#include <hip/hip_runtime.h>
#include <stdint.h>

// Problem constants (B=2, C=2, D=H=W=160, 10 skeleton iters, SMOOTH=1)
#define DIM   160
#define VOX   (DIM * DIM * DIM)     // 4,096,000 voxels per volume
#define NB    2
#define ITERS 10

// Padded volume layout: 2-voxel +inf guard band on every face.
// Guards give exact +inf padding for the erode min-pool AND make every
// TDM tile fully in-bounds (branch-free fixed-shape descriptors).
#define PDIM  164
#define PVOX  (PDIM * PDIM * PDIM)  // 4,410,944
#define NPVOL (NB * PVOX)           // 8,821,888

#define NTHREADS 256                // 8 wave32 per block
#define RB 1024                     // stage-1 reduction blocks per batch

// ---- fused open (erode + dilate + update) tiling -------------------------
// Output tile 32x8x8; skel staged with halo 2 (36x12x12) via one TDM load.
#define TX 32
#define TY 8
#define TZ 8
#define X0D 36
#define Y0D 12
#define Z0D 12
#define N_T0 (X0D * Y0D * Z0D)      // 5184  halo2 tile
#define N_TA (34 * 12 * 12)         // 4896  x-min
#define N_TB (34 * 10 * 12)         // 4080  xy-min
#define N_TC (34 * 10 * 10)         // 3400  erode tile (+ -inf rim mask)
#define N_TD (32 * 10 * 10)         // 3200  x-max of erode
#define N_TE (32 * 8 * 10)          // 2560  xy-max of erode
// Lifetime-overlapped LDS arena (floats):
//   T0 [0,5184) p1-p2 ; TA [5184,10080) p2-p3 ; TB [0,4080) p3-p4
//   TC [5184,8584) p4-end ; TD [0,3200) p5-p6 ; TE [8584,11144) p6-end
#define OFF_T0 0
#define OFF_TA 5184
#define OFF_TB 0
#define OFF_TC 5184
#define OFF_TD 0
#define OFF_TE 8584
#define SMEM_FLOATS 11144           // 44,576 bytes

struct F4 { float x, y, z, w; };

typedef unsigned int u32x4 __attribute__((ext_vector_type(4)));
typedef int i32x8 __attribute__((ext_vector_type(8)));
typedef int i32x4 __attribute__((ext_vector_type(4)));

__device__ __forceinline__ void wait_async0() {
#if __has_builtin(__builtin_amdgcn_s_wait_asynccnt)
  __builtin_amdgcn_s_wait_asynccnt(0);
#else
  asm volatile("s_wait_asynccnt 0" ::: "memory");
#endif
}

// One full soft-skeletonization iteration on the padded layout, fused:
//   minp = erode3(x)  (separable 3-tap min; +inf guards = SAME padding;
//                      out-of-volume erode rim forced to -inf)
//   dst  = relu(x - relu(dilate3(minp) - minp))   (separable 3-tap max)
__global__ void cldice_open_iter_kernel(const float* __restrict__ src,
                                        float* __restrict__ dst) {
  __shared__ float sm[SMEM_FLOATS];
  const int tid = threadIdx.x;
  int vol = blockIdx.z / (DIM / TZ);
  int tz  = blockIdx.z % (DIM / TZ);
  int X0 = blockIdx.x * TX, Y0 = blockIdx.y * TY, Z0 = tz * TZ;  // unpadded tile origin
  const float* vsrc = src + (size_t)vol * PVOX;
  float* vdst = dst + (size_t)vol * PVOX;

  // Phase 1: stage the 36x12x12 halo2 tile. Padded corner of the tile is
  // (X0, Y0, Z0) (= unpadded origin - 2 + guard 2), always fully in-bounds.
  float* T0 = sm + OFF_T0;
#if __has_builtin(__builtin_amdgcn_tensor_load_to_lds)
  if (tid < 32) {  // wave 0 issues one TDM descriptor for the whole tile
    unsigned lds0 = (unsigned)(size_t)T0;  // low 32 bits of flat addr = LDS offset
    unsigned long long gaddr = (unsigned long long)(size_t)vsrc +
        (unsigned long long)(((Z0 * PDIM + Y0) * PDIM + X0) * 4);
    // D# group 0: count=1 | lds_addr | global_addr[56:0] | type=2
    u32x4 g0 = { 1u,
                 lds0,
                 (unsigned)(gaddr & 0xffffffffu),
                 (unsigned)((gaddr >> 32) & 0x01ffffffu) | (2u << 30) };
    // D# group 1: data_size=4B; tensor_dim0/1=PDIM; tile 36x12x12;
    //             stride0=PDIM; stride1=PDIM^2
    i32x8 g1 = { (int)(2u << 16),
                 (int)(((unsigned)PDIM & 0xffffu) << 16),
                 (int)((unsigned)PDIM << 16),
                 (int)((unsigned)X0D << 16),
                 (int)((unsigned)Y0D | ((unsigned)Z0D << 16)),
                 (int)PDIM,
                 (int)((((unsigned)(PDIM * PDIM)) & 0xffffu) << 16),
                 (int)(((unsigned)(PDIM * PDIM)) >> 16) };
    // D# group 2: tensor_dim2=PDIM; stride2=PDIM^3; tile_dim3=0
    i32x4 g2 = { (int)PDIM, 0, (int)(PDIM * PDIM * PDIM), 0 };
    i32x4 g3 = { 0, 0, 0, 0 };
    i32x8 g4 = { 0, 0, 0, 0, 0, 0, 0, 0 };  // unused trailing group (clang-23 form)
    __builtin_amdgcn_tensor_load_to_lds(g0, g1, g2, g3, g4, 0);
    __builtin_amdgcn_s_wait_tensorcnt(0);
  }
  __syncthreads();
#else
  // Fallback: async global->LDS element copies (ASYNCcnt path)
  {
    unsigned lds0 = (unsigned)(size_t)T0;
    for (int i = tid; i < N_T0; i += NTHREADS) {
      int hx = i % X0D;
      int t  = i / X0D;
      int hy = t % Y0D;
      int hz = t / Y0D;
      unsigned boff = (unsigned)((((Z0 + hz) * PDIM + (Y0 + hy)) * PDIM + (X0 + hx)) * 4);
      unsigned loff = lds0 + (unsigned)(i * 4);
      asm volatile("global_load_async_to_lds_b32 %0, %1, %2"
                   :: "v"(loff), "v"(boff), "s"(vsrc)
                   : "memory");
    }
    wait_async0();
    __syncthreads();
  }
#endif

  // Grab this thread's 8 x-center values now; T0's arena is recycled later.
  float xc[8];
#pragma unroll
  for (int k = 0; k < 8; ++k) {
    int o = tid + k * NTHREADS;
    int ox = o & 31, oy = (o >> 5) & 7, oz = o >> 8;
    xc[k] = T0[((oz + 2) * Y0D + (oy + 2)) * X0D + (ox + 2)];
  }

  // Phase 2: x-min -> TA[34x12x12]
  float* TA = sm + OFF_TA;
  for (int i = tid; i < N_TA; i += NTHREADS) {
    int ix = i % 34; int t = i / 34; int iy = t % 12; int iz = t / 12;
    const float* r = &T0[(iz * Y0D + iy) * X0D + ix];
    TA[i] = fminf(fminf(r[0], r[1]), r[2]);
  }
  __syncthreads();

  // Phase 3: y-min -> TB[34x10x12]  (reuses T0 arena)
  float* TB = sm + OFF_TB;
  for (int i = tid; i < N_TB; i += NTHREADS) {
    int ix = i % 34; int t = i / 34; int iy = t % 10; int iz = t / 10;
    const float* r = &TA[(iz * 12 + iy) * 34 + ix];
    TB[i] = fminf(fminf(r[0], r[34]), r[68]);
  }
  __syncthreads();

  // Phase 4: z-min + out-of-volume mask -> erode tile TC[34x10x10]
  // Reference pads the eroded field with -inf before the max-pool.
  float* TC = sm + OFF_TC;
  for (int i = tid; i < N_TC; i += NTHREADS) {
    int ix = i % 34; int t = i / 34; int iy = t % 10; int iz = t / 10;
    const float* r = &TB[(iz * 10 + iy) * 34 + ix];
    float v = fminf(fminf(r[0], r[340]), r[680]);
    int gx = X0 - 1 + ix, gy = Y0 - 1 + iy, gz = Z0 - 1 + iz;  // unpadded coords
    bool valid = (unsigned)gx < DIM && (unsigned)gy < DIM && (unsigned)gz < DIM;
    TC[i] = valid ? v : -__builtin_inff();
  }
  __syncthreads();

  // Phase 5: x-max -> TD[32x10x10]  (reuses TB arena)
  float* TD = sm + OFF_TD;
  for (int i = tid; i < N_TD; i += NTHREADS) {
    int ox = i % 32; int t = i / 32; int iy = t % 10; int iz = t / 10;
    const float* r = &TC[(iz * 10 + iy) * 34 + ox];
    TD[i] = fmaxf(fmaxf(r[0], r[1]), r[2]);
  }
  __syncthreads();

  // Phase 6: y-max -> TE[32x8x10]
  float* TE = sm + OFF_TE;
  for (int i = tid; i < N_TE; i += NTHREADS) {
    int ox = i % 32; int t = i / 32; int oy = t % 8; int iz = t / 8;
    const float* r = &TD[(iz * 10 + oy) * 32 + ox];
    TE[i] = fmaxf(fmaxf(r[0], r[32]), r[64]);
  }
  __syncthreads();

  // Phase 7: z-max + combine + store (interior only -> guards persist)
#pragma unroll
  for (int k = 0; k < 8; ++k) {
    int o = tid + k * NTHREADS;
    int ox = o & 31, oy = (o >> 5) & 7, oz = o >> 8;
    const float* r = &TE[(oz * 8 + oy) * 32 + ox];
    float dil  = fmaxf(fmaxf(r[0], r[256]), r[512]);
    float minc = TC[((oz + 1) * 10 + (oy + 1)) * 34 + (ox + 1)];
    float contour = fmaxf(dil - minc, 0.0f);
    vdst[((Z0 + oz + 2) * PDIM + (Y0 + oy + 2)) * PDIM + (X0 + ox + 2)] =
        fmaxf(xc[k] - contour, 0.0f);
  }
}

// Build padded skeleton buffers: interior = softmax fg / onehot fg,
// guard band = +inf on all three ping-pong volumes (written every call).
__global__ void cldice_prep_kernel(const float* __restrict__ net,
                                   const int* __restrict__ tgt,
                                   float* __restrict__ sfg,
                                   float* __restrict__ stg,
                                   float* __restrict__ tmp) {
  int i = blockIdx.x * NTHREADS + threadIdx.x;
  if (i >= NPVOL) return;
  int vol = i / PVOX;
  int p = i - vol * PVOX;
  int pz = p / (PDIM * PDIM);
  int r  = p - pz * (PDIM * PDIM);
  int py = r / PDIM;
  int px = r - py * PDIM;
  int x = px - 2, y = py - 2, z = pz - 2;
  if ((unsigned)x < DIM && (unsigned)y < DIM && (unsigned)z < DIM) {
    int v = (z * DIM + y) * DIM + x;
    float a0 = net[(size_t)(vol * 2) * VOX + v];
    float a1 = net[(size_t)(vol * 2 + 1) * VOX + v];
    sfg[i] = 1.0f / (1.0f + expf(a0 - a1));   // softmax channel 1 of 2
    stg[i] = (tgt[(size_t)vol * VOX + v] == 1) ? 1.0f : 0.0f;
    tmp[i] = 0.0f;
  } else {
    float inf = __builtin_inff();
    sfg[i] = inf; stg[i] = inf; tmp[i] = inf;
  }
}

// Stage-1 reduction: per (batch, block) partial sums of
// {sum(sfg*tg), sum(sfg), sum(stg*fg), sum(stg)}; fg/tg recomputed on the fly
__global__ void cldice_reduce1_kernel(const float* __restrict__ net,
                                      const int* __restrict__ tgt,
                                      const float* __restrict__ sfg,
                                      const float* __restrict__ stg,
                                      F4* __restrict__ partials) {
  int b = blockIdx.y;
  const float* n0 = net + (size_t)(b * 2) * VOX;
  const float* n1 = n0 + VOX;
  const int* tb = tgt + (size_t)b * VOX;
  const float* sf = sfg + (size_t)b * PVOX;
  const float* st = stg + (size_t)b * PVOX;
  float a0 = 0.f, a1 = 0.f, a2 = 0.f, a3 = 0.f;
  for (int v = blockIdx.x * NTHREADS + threadIdx.x; v < VOX; v += RB * NTHREADS) {
    int z = v / (DIM * DIM);
    int rr = v - z * (DIM * DIM);
    int y = rr / DIM;
    int x = rr - y * DIM;
    int pidx = ((z + 2) * PDIM + (y + 2)) * PDIM + (x + 2);
    float f = 1.0f / (1.0f + expf(n0[v] - n1[v]));
    float t = (tb[v] == 1) ? 1.0f : 0.0f;
    float s1 = sf[pidx], s2 = st[pidx];
    a0 += s1 * t; a1 += s1; a2 += s2 * f; a3 += s2;
  }
  __shared__ float red[NTHREADS];
  float vals[4] = {a0, a1, a2, a3};
  F4 out;
  float* po = &out.x;
  for (int q = 0; q < 4; ++q) {
    red[threadIdx.x] = vals[q];
    __syncthreads();
    for (int s = NTHREADS / 2; s > 0; s >>= 1) {
      if ((int)threadIdx.x < s) red[threadIdx.x] += red[threadIdx.x + s];
      __syncthreads();
    }
    if (threadIdx.x == 0) po[q] = red[0];
    __syncthreads();
  }
  if (threadIdx.x == 0) partials[(size_t)b * RB + blockIdx.x] = out;
}

// Stage-2 reduction + scalar loss combine (single block, deterministic)
__global__ void cldice_reduce2_kernel(const F4* __restrict__ partials,
                                      float* __restrict__ out) {
  __shared__ F4 red[NTHREADS];
  __shared__ F4 sums[NB];
  for (int b = 0; b < NB; ++b) {
    F4 acc = {0.f, 0.f, 0.f, 0.f};
    for (int i = threadIdx.x; i < RB; i += NTHREADS) {
      F4 p = partials[(size_t)b * RB + i];
      acc.x += p.x; acc.y += p.y; acc.z += p.z; acc.w += p.w;
    }
    red[threadIdx.x] = acc;
    __syncthreads();
    for (int s = NTHREADS / 2; s > 0; s >>= 1) {
      if ((int)threadIdx.x < s) {
        red[threadIdx.x].x += red[threadIdx.x + s].x;
        red[threadIdx.x].y += red[threadIdx.x + s].y;
        red[threadIdx.x].z += red[threadIdx.x + s].z;
        red[threadIdx.x].w += red[threadIdx.x + s].w;
      }
      __syncthreads();
    }
    if (threadIdx.x == 0) sums[b] = red[0];
    __syncthreads();
  }
  if (threadIdx.x == 0) {
    float inter = 0.f, isum = 0.f, tsum = 0.f;
    for (int b = 0; b < NB; ++b) {
      float ifl = (sums[b].x + 1.0f) / (sums[b].y + 1.0f);  // norm_int(skel_fg, tg)
      float tfl = (sums[b].z + 1.0f) / (sums[b].w + 1.0f);  // norm_int(skel_tg, fg)
      inter += ifl * tfl; isum += ifl; tsum += tfl;
    }
    out[0] = -(2.0f * inter / (isum + tsum));
  }
}

extern "C" void kernel_launch(void* const* d_in, const int* in_sizes, int n_in,
                              void* d_out, int out_size, void* d_ws, size_t ws_size,
                              hipStream_t stream) {
  const float* net = (const float*)d_in[0];   // (B,C,D,H,W) f32
  const int* tgt   = (const int*)d_in[1];     // (B,1,D,H,W) int
  float* ws = (float*)d_ws;

  float* sfg = ws;                               // padded fg skeleton ping
  float* stg = ws + (size_t)NPVOL;               // padded tg skeleton ping
  float* tmp = ws + (size_t)2 * NPVOL;           // padded shared pong
  F4* partials = (F4*)(ws + (size_t)3 * NPVOL);  // NB*RB F4 partial sums
  float* out = (float*)d_out;

  cldice_prep_kernel<<<(NPVOL + NTHREADS - 1) / NTHREADS, NTHREADS, 0, stream>>>(
      net, tgt, sfg, stg, tmp);

  dim3 pg(DIM / TX, DIM / TY, (DIM / TZ) * NB);  // (5, 20, 40)
  // fg skeleton: ping-pong sfg <-> tmp; ITERS even -> result ends in sfg
  {
    float* a = sfg; float* b = tmp;
    for (int it = 0; it < ITERS; ++it) {
      cldice_open_iter_kernel<<<pg, NTHREADS, 0, stream>>>(a, b);
      float* t = a; a = b; b = t;
    }
  }
  // tg skeleton: ping-pong stg <-> tmp; result ends in stg
  {
    float* a = stg; float* b = tmp;
    for (int it = 0; it < ITERS; ++it) {
      cldice_open_iter_kernel<<<pg, NTHREADS, 0, stream>>>(a, b);
      float* t = a; a = b; b = t;
    }
  }

  dim3 rg(RB, NB);
  cldice_reduce1_kernel<<<rg, NTHREADS, 0, stream>>>(net, tgt, sfg, stg, partials);
  cldice_reduce2_kernel<<<1, NTHREADS, 0, stream>>>(partials, out);
}